// Transformer_79302276153378
// MI455X (gfx1250) — compile-verified
//
#include <hip/hip_runtime.h>
#include <hip/hip_bf16.h>

#define MUL 32
#define FOUT 128          // 4*MUL node feature width
#define ACCW 224          // edge_v / node_out width
#define HP 65             // LDS stride for 16x64 hidden tile
#define WP 129            // LDS stride for 16x128 weight-output tile
#define YP 33             // LDS stride for 16x32 Y tiles
#define INV_SQRT3 0.5773502691896258f

typedef __attribute__((ext_vector_type(2))) float v2f;
typedef __attribute__((ext_vector_type(8))) float v8f;

// D = A(16x4) * B(4x16) + C, fp32, wave32. 8-arg form:
// (neg_a, A, neg_b, B, c_mod, C, reuse_a, reuse_b)
__device__ __forceinline__ v8f wmma4(v2f a, v2f b, v8f c) {
  return __builtin_amdgcn_wmma_f32_16x16x4_f32(false, a, false, b, (short)0, c,
                                               false, false);
}

// tanh-form gelu rewritten as a sigmoid: 0.5*x*(1+tanh(u)) == x / (1+e^{-2u}).
// Branch-free: one v_exp_f32 + one v_rcp_f32 (vs the branchy libm tanhf).
__device__ __forceinline__ float gelu_f(float x) {
  const float k2 = 1.5957691216057308f; // 2*sqrt(2/pi)
  float u2 = k2 * (x + 0.044715f * x * x * x);
  return x * __frcp_rn(1.0f + __expf(-u2));
}

// 16-edge tile MLP: out = gelu(attr @ w1 / sqrt(8)) @ w2 / 8, fully via WMMA.
// Result (16 x 128) left in LDS tile W (stride WP). Ends with a barrier.
__device__ __forceinline__ void mlp16(const float* __restrict__ attr, int e0,
                                      int E, const float* __restrict__ w1,
                                      const float* __restrict__ w2,
                                      float* __restrict__ H,
                                      float* __restrict__ W, int lane) {
  const int half = lane >> 4, ln = lane & 15;
  const int er = min(e0 + ln, E - 1);
  // First GEMM: (16x8) @ (8x64).  K=8 -> 2 wmma steps.
  v2f af0, af1;
  {
    int kr = 2 * half;
    af0.x = attr[er * 8 + kr];
    af0.y = attr[er * 8 + kr + 1];
    kr = 4 + 2 * half;
    af1.x = attr[er * 8 + kr];
    af1.y = attr[er * 8 + kr + 1];
  }
  for (int nt = 0; nt < 4; ++nt) {
    v8f d = {};
    {
      int kr = 2 * half;
      v2f b;
      b.x = w1[kr * 64 + nt * 16 + ln];
      b.y = w1[(kr + 1) * 64 + nt * 16 + ln];
      d = wmma4(af0, b, d);
      kr = 4 + 2 * half;
      b.x = w1[kr * 64 + nt * 16 + ln];
      b.y = w1[(kr + 1) * 64 + nt * 16 + ln];
      d = wmma4(af1, b, d);
    }
#pragma unroll
    for (int r = 0; r < 8; ++r)
      H[(r + 8 * half) * HP + nt * 16 + ln] =
          gelu_f(d[r] * 0.35355339059327373f); // 1/sqrt(8)
  }
  __syncthreads();
  // Second GEMM: (16x64) @ (64x128).  K=64 -> 16 wmma steps, 8 N-tiles.
  for (int nt = 0; nt < 8; ++nt) {
    v8f d = {};
    for (int ks = 0; ks < 16; ++ks) {
      int kr = ks * 4 + 2 * half;
      v2f a;
      a.x = H[ln * HP + kr];
      a.y = H[ln * HP + kr + 1];
      v2f b;
      b.x = w2[kr * FOUT + nt * 16 + ln];
      b.y = w2[(kr + 1) * FOUT + nt * 16 + ln];
      d = wmma4(a, b, d);
    }
#pragma unroll
    for (int r = 0; r < 8; ++r)
      W[(r + 8 * half) * WP + nt * 16 + ln] = d[r] * 0.125f; // 1/sqrt(64)
  }
  __syncthreads();
}

// ---------------- Phase A: logits + exp + segment z ----------------
__global__ __launch_bounds__(32) void edge_logits_kernel(
    const int* __restrict__ esrc, const int* __restrict__ edst,
    const float* __restrict__ attr, const float* __restrict__ eattr,
    const float* __restrict__ cutoff, const float* __restrict__ nodef,
    const float* __restrict__ kw1, const float* __restrict__ kw2,
    const float* __restrict__ wss, const float* __restrict__ wvv,
    float* __restrict__ expo, float* __restrict__ zacc, int E) {
  __shared__ float H[16 * HP];
  __shared__ float Wk[16 * WP];
  __shared__ float Xd[16 * YP];
  __shared__ float Y1[16 * YP], Y2[16 * YP], Y3[16 * YP];
  __shared__ float Y40[16 * YP], Y41[16 * YP], Y42[16 * YP];
  __shared__ int lSrc[16], lDst[16];
  __shared__ float lA0[16], lA1x[16], lA1y[16], lA1z[16];
  __shared__ float lLogit[16];

  const int lane = threadIdx.x;
  const int half = lane >> 4, ln = lane & 15;
  const int e0 = blockIdx.x * 16;

  if (lane < 16) {
    int er = min(e0 + lane, E - 1);
    lSrc[lane] = esrc[er];
    lDst[lane] = edst[er];
    lA0[lane] = eattr[er * 4 + 0];
    lA1x[lane] = eattr[er * 4 + 1];
    lA1y[lane] = eattr[er * 4 + 2];
    lA1z[lane] = eattr[er * 4 + 3];
  }
  mlp16(attr, e0, E, kw1, kw2, H, Wk, lane); // Wk = w_k tile, barrier inside

  // Build Y tiles (elementwise tensor-product factors), one edge per iter,
  // one u-channel per lane.  node_f is L2-resident (10 MB).
  for (int e = 0; e < 16; ++e) {
    const int u = lane;
    const int se = lSrc[e], de = lDst[e];
    const float a0 = lA0[e], ax = lA1x[e], ay = lA1y[e], az = lA1z[e];
    const float* ns = nodef + (size_t)se * FOUT;
    const float* nd = nodef + (size_t)de * FOUT;
    float s = ns[u];
    float vs0 = ns[32 + u * 3 + 0], vs1 = ns[32 + u * 3 + 1],
          vs2 = ns[32 + u * 3 + 2];
    float vd0 = nd[32 + u * 3 + 0], vd1 = nd[32 + u * 3 + 1],
          vd2 = nd[32 + u * 3 + 2];
    float w0 = Wk[e * WP + u], w1v = Wk[e * WP + 32 + u],
          w2v = Wk[e * WP + 64 + u], w3v = Wk[e * WP + 96 + u];
    float vsdot = vs0 * ax + vs1 * ay + vs2 * az;
    Y1[e * YP + u] = w0 * s * a0;                 // ks0
    Y2[e * YP + u] = w3v * vsdot * INV_SQRT3;     // ks1
    Y3[e * YP + u] = INV_SQRT3 * w1v * s;         // (w1*s) factor of kv0 term
    float w2s = INV_SQRT3 * a0 * w2v;
    Y40[e * YP + u] = w2s * vs0;                  // kv1 per k-component
    Y41[e * YP + u] = w2s * vs1;
    Y42[e * YP + u] = w2s * vs2;
    Xd[e * YP + u] = vd0 * ax + vd1 * ay + vd2 * az; // sum_k v_dst*a1
  }
  __syncthreads();

  const int deL = lDst[ln]; // this lane's A-fragment row (edge ln) dst node
  const float* ndL = nodef + (size_t)deL * FOUT;

  float accd[8];
#pragma unroll
  for (int r = 0; r < 8; ++r) accd[r] = 0.0f;

  // Z = X @ W (16x32 @ 32x32 via WMMA), then accumulate Z .* Y per edge.
  auto gemm_dot = [&](auto getA, const float* __restrict__ W,
                      const float* __restrict__ Y) {
    for (int nt = 0; nt < 2; ++nt) {
      v8f d = {};
#pragma unroll
      for (int ks = 0; ks < 8; ++ks) {
        int kr = ks * 4 + 2 * half;
        v2f a = getA(kr);
        v2f b;
        b.x = W[kr * MUL + nt * 16 + ln];
        b.y = W[(kr + 1) * MUL + nt * 16 + ln];
        d = wmma4(a, b, d);
      }
#pragma unroll
      for (int r = 0; r < 8; ++r)
        accd[r] += d[r] * Y[(r + 8 * half) * YP + nt * 16 + ln];
    }
  };

  auto A_sdst = [&](int kr) {
    v2f a; a.x = ndL[kr]; a.y = ndL[kr + 1]; return a;
  };
  auto A_xd = [&](int kr) {
    v2f a; a.x = Xd[ln * YP + kr]; a.y = Xd[ln * YP + kr + 1]; return a;
  };

  gemm_dot(A_sdst, wss, Y1);             // s_dst^T Wss0 ks0
  gemm_dot(A_sdst, wss + MUL * MUL, Y2); // s_dst^T Wss1 ks1
  gemm_dot(A_xd, wvv, Y3);               // (v_dst.a1)^T Wvv0 (w1*s_src)
  for (int k = 0; k < 3; ++k) {
    auto A_vd = [&](int kr) {
      v2f a;
      a.x = ndL[32 + kr * 3 + k];
      a.y = ndL[32 + (kr + 1) * 3 + k];
      return a;
    };
    const float* Yk = (k == 0) ? Y40 : (k == 1) ? Y41 : Y42;
    gemm_dot(A_vd, wvv + MUL * MUL, Yk); // v_dst_k^T Wvv1 (a0*w2*v_src_k)
  }

  // Reduce over columns: sum across the 16 lanes of each half-wave.
#pragma unroll
  for (int r = 0; r < 8; ++r) {
    float v = accd[r];
    v += __shfl_xor(v, 1, 32);
    v += __shfl_xor(v, 2, 32);
    v += __shfl_xor(v, 4, 32);
    v += __shfl_xor(v, 8, 32);
    accd[r] = v;
  }
  if (ln == 0) {
#pragma unroll
    for (int r = 0; r < 8; ++r) lLogit[r + 8 * half] = accd[r];
  }
  __syncthreads();

  if (lane < 16 && e0 + lane < E) {
    float lg = lLogit[lane] * (1.0f / 64.0f);
    float ex = cutoff[e0 + lane] * __expf(lg);
    expo[e0 + lane] = ex;
    unsafeAtomicAdd(&zacc[lDst[lane]], ex);
  }
}

// ---------------- Phase B: alpha, edge_v, scatter-add ----------------
__global__ __launch_bounds__(32) void edge_scatter_kernel(
    const int* __restrict__ esrc, const int* __restrict__ edst,
    const float* __restrict__ attr, const float* __restrict__ eattr,
    const float* __restrict__ nodef, const float* __restrict__ vw1,
    const float* __restrict__ vw2, const float* __restrict__ expo,
    const float* __restrict__ zacc, float* __restrict__ acc, int E) {
  __shared__ float H[16 * HP];
  __shared__ float Wv[16 * WP];
  __shared__ int lSrc[16], lDst[16];
  __shared__ float lA0[16], lA1x[16], lA1y[16], lA1z[16], lCoef[16];

  const int lane = threadIdx.x;
  const int e0 = blockIdx.x * 16;

  if (lane < 16) {
    int er = min(e0 + lane, E - 1);
    lSrc[lane] = esrc[er];
    lDst[lane] = edst[er];
    lA0[lane] = eattr[er * 4 + 0];
    lA1x[lane] = eattr[er * 4 + 1];
    lA1y[lane] = eattr[er * 4 + 2];
    lA1z[lane] = eattr[er * 4 + 3];
    float zz = zacc[edst[er]];
    if (zz == 0.0f) zz = 1.0f;
    float al = expo[er] / zz;
    lCoef[lane] = sqrtf(fmaxf(al, 0.0f)); // sqrt(relu(alpha))
  }
  mlp16(attr, e0, E, vw1, vw2, H, Wv, lane); // Wv = w_v tile, barrier inside

  for (int e = 0; e < 16; ++e) {
    if (e0 + e >= E) break; // uniform across wave
    const int u = lane;
    const int se = lSrc[e], de = lDst[e];
    const float a0 = lA0[e], ax = lA1x[e], ay = lA1y[e], az = lA1z[e];
    const float c = lCoef[e];
    const float* ns = nodef + (size_t)se * FOUT;
    float s = ns[u];
    float vs0 = ns[32 + u * 3 + 0], vs1 = ns[32 + u * 3 + 1],
          vs2 = ns[32 + u * 3 + 2];
    float w0 = Wv[e * WP + u], w1v = Wv[e * WP + 32 + u],
          w2v = Wv[e * WP + 64 + u], w3v = Wv[e * WP + 96 + u];
    float vsdot = vs0 * ax + vs1 * ay + vs2 * az;
    float* ab = acc + (size_t)de * ACCW;
    unsafeAtomicAdd(&ab[u], c * w0 * s * a0);                       // out_s0
    unsafeAtomicAdd(&ab[32 + u], c * w3v * vsdot * INV_SQRT3);      // out_s1
    float t0 = c * w1v * s;                                         // out_v0
    unsafeAtomicAdd(&ab[64 + u * 3 + 0], t0 * ax);
    unsafeAtomicAdd(&ab[64 + u * 3 + 1], t0 * ay);
    unsafeAtomicAdd(&ab[64 + u * 3 + 2], t0 * az);
    float t1 = c * w2v * a0;                                        // out_v1
    unsafeAtomicAdd(&ab[160 + u * 3 + 0], t1 * vs0);
    unsafeAtomicAdd(&ab[160 + u * 3 + 1], t1 * vs1);
    unsafeAtomicAdd(&ab[160 + u * 3 + 2], t1 * vs2);
  }
}

// ---------------- Phase C: per-node output linears ----------------
__global__ __launch_bounds__(32) void node_out_kernel(
    const float* __restrict__ acc, const float* __restrict__ lws,
    const float* __restrict__ lwv, float* __restrict__ out, int N) {
  const int lane = threadIdx.x;
  const int half = lane >> 4, ln = lane & 15;
  const int n0 = blockIdx.x * 16;
  const int nr = min(n0 + ln, N - 1);
  const float* arow = acc + (size_t)nr * ACCW;

  // out_s = sc(64) @ lin_ws(64x32) / 8
  for (int nt = 0; nt < 2; ++nt) {
    v8f d = {};
    for (int ks = 0; ks < 16; ++ks) {
      int kr = ks * 4 + 2 * half;
      v2f a; a.x = arow[kr]; a.y = arow[kr + 1];
      v2f b;
      b.x = lws[kr * MUL + nt * 16 + ln];
      b.y = lws[(kr + 1) * MUL + nt * 16 + ln];
      d = wmma4(a, b, d);
    }
#pragma unroll
    for (int r = 0; r < 8; ++r) {
      int m = n0 + r + 8 * half;
      if (m < N) out[(size_t)m * FOUT + nt * 16 + ln] = d[r] * 0.125f;
    }
  }
  // out_v[:, :, k] = vc[:, :, k](64) @ lin_wv(64x32) / 8  for k=0..2
  for (int k = 0; k < 3; ++k) {
    for (int nt = 0; nt < 2; ++nt) {
      v8f d = {};
      for (int ks = 0; ks < 16; ++ks) {
        int kr = ks * 4 + 2 * half;
        v2f a;
        a.x = arow[64 + kr * 3 + k];
        a.y = arow[64 + (kr + 1) * 3 + k];
        v2f b;
        b.x = lwv[kr * MUL + nt * 16 + ln];
        b.y = lwv[(kr + 1) * MUL + nt * 16 + ln];
        d = wmma4(a, b, d);
      }
#pragma unroll
      for (int r = 0; r < 8; ++r) {
        int m = n0 + r + 8 * half;
        if (m < N)
          out[(size_t)m * FOUT + 32 + (nt * 16 + ln) * 3 + k] = d[r] * 0.125f;
      }
    }
  }
}

extern "C" void kernel_launch(void* const* d_in, const int* in_sizes, int n_in,
                              void* d_out, int out_size, void* d_ws,
                              size_t ws_size, hipStream_t stream) {
  const int* esrc = (const int*)d_in[0];
  const int* edst = (const int*)d_in[1];
  const float* attr = (const float*)d_in[2];
  const float* eattr = (const float*)d_in[3];
  const float* cutoff = (const float*)d_in[4];
  const float* nodef = (const float*)d_in[5];
  const float* kw1 = (const float*)d_in[6];
  const float* kw2 = (const float*)d_in[7];
  const float* vw1 = (const float*)d_in[8];
  const float* vw2 = (const float*)d_in[9];
  const float* wss = (const float*)d_in[10];
  const float* wvv = (const float*)d_in[11];
  const float* lws = (const float*)d_in[12];
  const float* lwv = (const float*)d_in[13];

  const int E = in_sizes[0];
  const int N = in_sizes[5] / FOUT;

  float* ws = (float*)d_ws;
  float* expo = ws;            // E floats
  float* zacc = expo + E;      // N floats
  float* acc = zacc + N;       // N*224 floats  (total ~20 MB)

  hipMemsetAsync(zacc, 0, (size_t)N * sizeof(float), stream);
  hipMemsetAsync(acc, 0, (size_t)N * ACCW * sizeof(float), stream);

  const int etiles = (E + 15) / 16;
  const int ntiles = (N + 15) / 16;
  edge_logits_kernel<<<etiles, 32, 0, stream>>>(esrc, edst, attr, eattr, cutoff,
                                                nodef, kw1, kw2, wss, wvv, expo,
                                                zacc, E);
  edge_scatter_kernel<<<etiles, 32, 0, stream>>>(esrc, edst, attr, eattr, nodef,
                                                 vw1, vw2, expo, zacc, acc, E);
  node_out_kernel<<<ntiles, 32, 0, stream>>>(acc, lws, lwv, (float*)d_out, N);
}